// GatedGCNNet_26474178413024
// MI455X (gfx1250) — compile-verified
//
#include <hip/hip_runtime.h>
#include <hip/hip_bf16.h>

#define H 128
#define EPS_BN  1e-5f
#define EPS_DEN 1e-6f

// LDS staging of W in KC-row slabs, stored pair-interleaved:
//   element (k, j) -> lds_w[(k>>1)*LDSP + 2*j + (k&1)]
// so a WMMA B-fragment (k, k+1 at column j) is one contiguous 8-byte load.
// Pair-row stride LDSP = 2*H + 32 = 288 floats: lanes 0-15 hit banks 0..31,
// lanes 16-31 (+1 pair-row, 288 % 64 == 32) hit banks 32..63 -> conflict-free.
#define KC 64
#define LDSP 288
#define LDS_FLOATS ((KC / 2) * LDSP)

typedef __attribute__((ext_vector_type(2))) float v2f;
typedef __attribute__((ext_vector_type(8))) float v8f;

__device__ __forceinline__ float atomic_add_f32(float* p, float v) {
  return __hip_atomic_fetch_add(p, v, __ATOMIC_RELAXED, __HIP_MEMORY_SCOPE_AGENT);
}

// 8 independent WMMA chains over the 8 column tiles of a 16x128 row-block.
__device__ __forceinline__ void wmma8(const v2f a, const float* wrow,
                                      v8f acc[8]) {
#pragma unroll
  for (int t = 0; t < 8; ++t) {
    const v2f b = *(const v2f*)(wrow + t * 32);  // ds_load_b64, pre-paired
    acc[t] = __builtin_amdgcn_wmma_f32_16x16x4_f32(
        false, a, false, b, (short)0, acc[t], false, false);
  }
}

// ---------------------------------------------------------------------------
// Core: one wave accumulates a 16x128 row-block of  A[M x K] @ W[K x 128].
// Requires: K % 4 == 0, M % 64 == 0, blockDim.x == 128 (4 waves).
// Layouts per CDNA5 ISA 7.12.2:
//   A 16x4 f32 frag : lanes 0-15 -> M=0..15 (K=0,1), lanes 16-31 -> (K=2,3)
//   C/D 16x16 f32   : VGPR v, lanes 0-15 -> M=v, lanes 16-31 -> M=v+8
// ---------------------------------------------------------------------------
__device__ __forceinline__ void gemm_core(const float* __restrict__ A,
                                          const float* __restrict__ W,
                                          int K, int m0, int half, int col,
                                          float* lds_w, v8f acc[8]) {
  const float* Arow = A + (size_t)(m0 + col) * K;  // lane's A-fragment row
  for (int kc = 0; kc < K; kc += KC) {
    const int klen = (K - kc < KC) ? (K - kc) : KC;
    __syncthreads();  // protect LDS against previous-slab readers
    // Stage pair-interleaved W slab: two float2 global loads -> one b128 store.
    for (int i = threadIdx.x; i < (klen >> 1) * (H >> 1); i += 128) {
      const int r = i >> 6;         // pair-row within slab
      const int j = (i & 63) << 1;  // even column
      const float2 w0 = *(const float2*)(W + (size_t)(kc + 2 * r) * H + j);
      const float2 w1 = *(const float2*)(W + (size_t)(kc + 2 * r + 1) * H + j);
      const float4 q = make_float4(w0.x, w1.x, w0.y, w1.y);
      *(float4*)(&lds_w[r * LDSP + 2 * j]) = q;
    }
    __syncthreads();

    // Speculative prefetch of the next W slab (uniform branch; overlaps WMMA).
    if (kc + KC < K)
      __builtin_prefetch(W + (size_t)(kc + KC) * H + (threadIdx.x << 5), 0, 0);

    // Software-pipelined A fragments: load one k-step ahead so the b64 load
    // latency hides under the 8-WMMA octet.
    v2f a = *(const v2f*)(Arow + kc + 2 * half);
    int k = 0;
    for (; k + 4 < klen; k += 4) {
      const v2f an = *(const v2f*)(Arow + kc + k + 4 + 2 * half);
      wmma8(a, &lds_w[((k >> 1) + half) * LDSP + 2 * col], acc);
      a = an;
    }
    wmma8(a, &lds_w[((k >> 1) + half) * LDSP + 2 * col], acc);
  }
}

// C[M x 128] = A[M x K] @ W[K x 128] + bias, optional ReLU.
__global__ __launch_bounds__(128) void gemm128_kernel(
    const float* __restrict__ A, const float* __restrict__ W,
    const float* __restrict__ bias, float* __restrict__ C,
    int M, int K, int relu) {
  __shared__ float lds_w[LDS_FLOATS];
  const int wave = threadIdx.x >> 5;
  const int lane = threadIdx.x & 31;
  const int half = lane >> 4;
  const int col  = lane & 15;
  const int m0 = (blockIdx.x * 4 + wave) * 16;

  v8f acc[8] = {};
  gemm_core(A, W, K, m0, half, col, lds_w, acc);

#pragma unroll
  for (int t = 0; t < 8; ++t) {
    const int j = t * 16 + col;
    const float bb = bias[j];
#pragma unroll
    for (int v = 0; v < 8; ++v) {
      const int row = m0 + v + 8 * half;
      float val = acc[t][v] + bb;
      if (relu) val = fmaxf(val, 0.f);
      C[(size_t)row * H + j] = val;
    }
  }
}

// Fused edge update: e_new = e@Wc + bc + Dh[src] + Eh[dst]; sigma = sigmoid;
// scatter num += sigma*Bh[src], den += sigma (hw f32 atomics). One pass over e.
__global__ __launch_bounds__(128) void edge_layer_kernel(
    const float* __restrict__ Ein, const float* __restrict__ W,
    const float* __restrict__ bias,
    const float* __restrict__ Dh, const float* __restrict__ Ehm,
    const float* __restrict__ Bh,
    const int* __restrict__ src, const int* __restrict__ dst,
    float* __restrict__ Eout, float* __restrict__ num, float* __restrict__ den,
    int E) {
  __shared__ float lds_w[LDS_FLOATS];
  const int wave = threadIdx.x >> 5;
  const int lane = threadIdx.x & 31;
  const int half = lane >> 4;
  const int col  = lane & 15;
  const int m0 = (blockIdx.x * 4 + wave) * 16;

  v8f acc[8] = {};
  gemm_core(Ein, W, H, m0, half, col, lds_w, acc);

#pragma unroll
  for (int v = 0; v < 8; ++v) {
    const int em = m0 + v + 8 * half;
    const int s = src[em];
    const int d = dst[em];
#pragma unroll
    for (int t = 0; t < 8; ++t) {
      const int j = t * 16 + col;
      float enew = acc[t][v] + bias[j] + Dh[(size_t)s * H + j] +
                   Ehm[(size_t)d * H + j];
      Eout[(size_t)em * H + j] = enew;
      const float sig = 1.f / (1.f + __expf(-enew));
      atomic_add_f32(&num[(size_t)d * H + j], sig * Bh[(size_t)s * H + j]);
      atomic_add_f32(&den[(size_t)d * H + j], sig);
    }
  }
}

// Edge encoder: K=2, pure elementwise (coalesced).
__global__ void encode_edges_kernel(const float* __restrict__ ef,
                                    const float* __restrict__ w,
                                    const float* __restrict__ b,
                                    float* __restrict__ e, long tot) {
  const long idx = (long)blockIdx.x * blockDim.x + threadIdx.x;
  if (idx >= tot) return;
  const long m = idx >> 7;
  const int j = (int)(idx & (H - 1));
  e[idx] = ef[2 * m] * w[j] + ef[2 * m + 1] * w[H + j] + b[j];
}

// h_new = Ah + num/(den+eps)  (written in-place into `num`)
__global__ void combine_kernel(const float* __restrict__ Ah,
                               const float* __restrict__ num,
                               const float* __restrict__ den,
                               float* __restrict__ out, long tot) {
  const long idx = (long)blockIdx.x * blockDim.x + threadIdx.x;
  if (idx >= tot) return;
  out[idx] = Ah[idx] + num[idx] / (den[idx] + EPS_DEN);
}

// Column-wise sum / sum-of-squares partials (coalesced rows, atomic merge).
__global__ void bn_stats_kernel(const float* __restrict__ x, int rows,
                                float* __restrict__ sum,
                                float* __restrict__ sumsq) {
  const int j = threadIdx.x;  // 0..127
  float s = 0.f, q = 0.f;
  for (int i = blockIdx.x; i < rows; i += gridDim.x) {
    const float v = x[(size_t)i * H + j];
    s += v;
    q += v * v;
  }
  atomic_add_f32(&sum[j], s);
  atomic_add_f32(&sumsq[j], q);
}

__global__ void bn_finalize_kernel(const float* __restrict__ sum,
                                   const float* __restrict__ sumsq,
                                   const float* __restrict__ g,
                                   const float* __restrict__ b, int rows,
                                   float* __restrict__ scale,
                                   float* __restrict__ shift) {
  const int j = threadIdx.x;
  const float inv = 1.f / (float)rows;
  const float m = sum[j] * inv;
  const float v = sumsq[j] * inv - m * m;
  const float sc = g[j] * rsqrtf(v + EPS_BN);
  scale[j] = sc;
  shift[j] = b[j] - m * sc;
}

__global__ void bn_apply_relu_kernel(const float* __restrict__ x,
                                     const float* __restrict__ sc,
                                     const float* __restrict__ sh,
                                     float* __restrict__ y, long tot) {
  const long idx = (long)blockIdx.x * blockDim.x + threadIdx.x;
  if (idx >= tot) return;
  const int j = (int)(idx & (H - 1));
  y[idx] = fmaxf(x[idx] * sc[j] + sh[j], 0.f);
}

// Readout second layer: [N,128] @ [128,NC] + b  (tiny NC -> VALU dot).
__global__ void mlp2_kernel(const float* __restrict__ x,
                            const float* __restrict__ w,
                            const float* __restrict__ b,
                            float* __restrict__ out, int n, int nc) {
  const long idx = (long)blockIdx.x * blockDim.x + threadIdx.x;
  if (idx >= (long)n * nc) return;
  const int i = (int)(idx / nc);
  const int c = (int)(idx % nc);
  float acc = b[c];
  for (int k = 0; k < H; ++k) acc += x[(size_t)i * H + k] * w[k * nc + c];
  out[idx] = acc;
}

extern "C" void kernel_launch(void* const* d_in, const int* in_sizes, int n_in,
                              void* d_out, int out_size, void* d_ws,
                              size_t ws_size, hipStream_t stream) {
  const float* node_feat = (const float*)d_in[0];
  const float* edge_feat = (const float*)d_in[1];
  const int*   src       = (const int*)d_in[2];
  const int*   dst       = (const int*)d_in[3];
  const float* enc_nw    = (const float*)d_in[4];
  const float* enc_nb    = (const float*)d_in[5];
  const float* enc_ew    = (const float*)d_in[6];
  const float* enc_eb    = (const float*)d_in[7];
  const float* lin_w     = (const float*)d_in[8];
  const float* lin_b     = (const float*)d_in[9];
  const float* bnh_g     = (const float*)d_in[10];
  const float* bnh_b     = (const float*)d_in[11];
  const float* bne_g     = (const float*)d_in[12];
  const float* bne_b     = (const float*)d_in[13];
  const float* mlp_w1    = (const float*)d_in[14];
  const float* mlp_b1    = (const float*)d_in[15];
  const float* mlp_w2    = (const float*)d_in[16];
  const float* mlp_b2    = (const float*)d_in[17];

  const int IN_N = in_sizes[4] / H;            // 776
  const int Nn   = in_sizes[0] / IN_N;         // 40000
  const int E    = in_sizes[2];                // 640000
  const int L    = in_sizes[8] / (5 * H * H);  // 2
  const int NC   = in_sizes[16] / H;           // 5

  const size_t NH = (size_t)Nn * H;
  const size_t EH = (size_t)E * H;

  // Workspace carve-out
  float* h0  = (float*)d_ws;
  float* Ah  = h0 + NH;
  float* Bh  = Ah + NH;
  float* Dh  = Bh + NH;
  float* Eh  = Dh + NH;
  float* num = Eh + NH;
  float* den = num + NH;
  float* e0  = den + NH;
  float* e1  = e0 + EH;
  float* st  = e1 + EH;  // 4*H floats: sum, sumsq, scale, shift

  const dim3 blk(128);
  const dim3 gN(Nn / 64);  // Nn % 64 == 0
  const dim3 gE(E / 64);   // E  % 64 == 0

  // Encoders
  gemm128_kernel<<<gN, blk, 0, stream>>>(node_feat, enc_nw, enc_nb, h0, Nn,
                                         IN_N, 0);
  {
    const long tot = (long)EH;
    encode_edges_kernel<<<(int)((tot + 255) / 256), 256, 0, stream>>>(
        edge_feat, enc_ew, enc_eb, e0, tot);
  }

  for (int l = 0; l < L; ++l) {
    const float* lw = lin_w + (size_t)l * 5 * H * H;
    const float* lb = lin_b + (size_t)l * 5 * H;

    // A, B, D, E node linears (WMMA)
    gemm128_kernel<<<gN, blk, 0, stream>>>(h0, lw + 0 * H * H, lb + 0 * H, Ah, Nn, H, 0);
    gemm128_kernel<<<gN, blk, 0, stream>>>(h0, lw + 1 * H * H, lb + 1 * H, Bh, Nn, H, 0);
    gemm128_kernel<<<gN, blk, 0, stream>>>(h0, lw + 3 * H * H, lb + 3 * H, Dh, Nn, H, 0);
    gemm128_kernel<<<gN, blk, 0, stream>>>(h0, lw + 4 * H * H, lb + 4 * H, Eh, Nn, H, 0);

    hipMemsetAsync(num, 0, NH * sizeof(float), stream);
    hipMemsetAsync(den, 0, NH * sizeof(float), stream);

    // Fused: Ce GEMM + gather + sigmoid + scatter atomics, writes e_new -> e1
    edge_layer_kernel<<<gE, blk, 0, stream>>>(e0, lw + 2 * H * H, lb + 2 * H,
                                              Dh, Eh, Bh, src, dst, e1, num,
                                              den, E);

    // h_new = Ah + num/(den+eps)  (in-place into num)
    {
      const long tot = (long)NH;
      combine_kernel<<<(int)((tot + 255) / 256), 256, 0, stream>>>(Ah, num, den,
                                                                   num, tot);
    }

    // BN(h_new) + ReLU -> h0
    hipMemsetAsync(st, 0, 2 * H * sizeof(float), stream);
    bn_stats_kernel<<<512, H, 0, stream>>>(num, Nn, st, st + H);
    bn_finalize_kernel<<<1, H, 0, stream>>>(st, st + H, bnh_g + l * H,
                                            bnh_b + l * H, Nn, st + 2 * H,
                                            st + 3 * H);
    {
      const long tot = (long)NH;
      bn_apply_relu_kernel<<<(int)((tot + 255) / 256), 256, 0, stream>>>(
          num, st + 2 * H, st + 3 * H, h0, tot);
    }

    // BN(e_new) + ReLU -> e0 (becomes next layer's edge input)
    hipMemsetAsync(st, 0, 2 * H * sizeof(float), stream);
    bn_stats_kernel<<<1024, H, 0, stream>>>(e1, E, st, st + H);
    bn_finalize_kernel<<<1, H, 0, stream>>>(st, st + H, bne_g + l * H,
                                            bne_b + l * H, E, st + 2 * H,
                                            st + 3 * H);
    {
      const long tot = (long)EH;
      bn_apply_relu_kernel<<<(int)((tot + 255) / 256), 256, 0, stream>>>(
          e1, st + 2 * H, st + 3 * H, e0, tot);
    }
  }

  // Readout: relu(h @ w1 + b1) (WMMA) then small [128,NC] dot.
  gemm128_kernel<<<gN, blk, 0, stream>>>(h0, mlp_w1, mlp_b1, Ah, Nn, H, 1);
  {
    const long tot = (long)Nn * NC;
    mlp2_kernel<<<(int)((tot + 255) / 256), 256, 0, stream>>>(
        Ah, mlp_w2, mlp_b2, (float*)d_out, Nn, NC);
  }
}